// GIN_23390391894890
// MI455X (gfx1250) — compile-verified
//
#include <hip/hip_runtime.h>

typedef float v2f __attribute__((ext_vector_type(2)));
typedef float v8f __attribute__((ext_vector_type(8)));

#define HDIM 128
#define LDA 132   // padded LDS stride (floats) to dodge bank conflicts

// ---------------------------------------------------------------- copy (float4)
__global__ __launch_bounds__(256) void copy_f4(const float4* __restrict__ in,
                                               float4* __restrict__ out, int n4) {
  int i = blockIdx.x * 256 + threadIdx.x;
  if (i < n4) out[i] = in[i];
}

__global__ __launch_bounds__(256) void zero_f(float* p, int n) {
  int i = blockIdx.x * 256 + threadIdx.x;
  if (i < n) p[i] = 0.0f;
}

// ---------------------------------------------------------------- edge scatter
// out[dst] += feat[src]; one 32-lane group per edge, float4 per lane.
__global__ __launch_bounds__(256) void scatter_add(const float* __restrict__ feat,
                                                   const int* __restrict__ ei,
                                                   float* __restrict__ out, int E) {
  int gid = blockIdx.x * 256 + threadIdx.x;
  int e = gid >> 5;
  if (e >= E) return;
  int q = gid & 31;
  int s = ei[e];
  int d = ei[E + e];
  const float4 v = ((const float4*)(feat + (size_t)s * HDIM))[q];
  float* o = out + (size_t)d * HDIM + q * 4;
  atomicAdd(o + 0, v.x);
  atomicAdd(o + 1, v.y);
  atomicAdd(o + 2, v.z);
  atomicAdd(o + 3, v.w);
}

// ---------------------------------------------------------------- WMMA GEMM
// out[N x 128] = epilogue(A[N x 128] @ W[128 x 128] + bias)
// mode 0: relu(t) ; mode 1: relu(bn(t)) ; mode 2: bn(relu(t))
// Block: 256 thr (8 waves). 64 rows x 128 cols per block.
// Wave w: row tile (w>>1), col tiles ((w&1)*4 .. +3). K stepped by 4 via
// V_WMMA_F32_16X16X4_F32.
__global__ __launch_bounds__(256) void gemm128_wmma(
    const float* __restrict__ A, const float* __restrict__ W,
    const float* __restrict__ bias, const float* __restrict__ g,
    const float* __restrict__ be, const float* __restrict__ m,
    const float* __restrict__ v, float* __restrict__ out, int N, int mode) {
  __shared__ float Alds[64][LDA];   // 64 rows of A
  __shared__ float Wt[128][LDA];    // W transposed: Wt[n][k] = W[k][n]

  const int tid = threadIdx.x;
  const int baseRow = blockIdx.x * 64;

  // Stage A tile (zero-padded past N)
  for (int i = tid; i < 64 * 32; i += 256) {
    int r = i >> 5, c4 = i & 31;
    float4 f4 = make_float4(0.0f, 0.0f, 0.0f, 0.0f);
    int grow = baseRow + r;
    if (grow < N) f4 = ((const float4*)(A + (size_t)grow * HDIM))[c4];
    *(float4*)&Alds[r][c4 * 4] = f4;
  }
  // Stage W transposed (coalesced global read, strided LDS write)
  for (int i = tid; i < 128 * 32; i += 256) {
    int k = i >> 5, n4 = i & 31;
    float4 w4 = ((const float4*)(W + (size_t)k * HDIM))[n4];
    Wt[n4 * 4 + 0][k] = w4.x;
    Wt[n4 * 4 + 1][k] = w4.y;
    Wt[n4 * 4 + 2][k] = w4.z;
    Wt[n4 * 4 + 3][k] = w4.w;
  }
  __syncthreads();

  const int lane = tid & 31;
  const int wave = tid >> 5;
  const int rtile = wave >> 1;           // 0..3
  const int cg = (wave & 1) * 4;         // col-tile base: 0 or 4
  const int mrow = lane & 15;            // M (for A) / N (for B) within tile
  const int khalf = (lane >> 4) << 1;    // lanes 16-31 hold K+2,K+3

  v8f acc[4] = {};
  const int arow = rtile * 16 + mrow;

  for (int k = 0; k < 128; k += 4) {
    const int kb = k + khalf;
    v2f a = *(const v2f*)&Alds[arow][kb];
#pragma unroll
    for (int t = 0; t < 4; ++t) {
      v2f b = *(const v2f*)&Wt[(cg + t) * 16 + mrow][kb];
      acc[t] = __builtin_amdgcn_wmma_f32_16x16x4_f32(
          /*neg_a=*/false, a, /*neg_b=*/false, b,
          /*c_mod=*/(short)0, acc[t], /*reuse_a=*/false, /*reuse_b=*/false);
    }
  }

  const int hi = lane >> 4;  // 0: rows r..; 1: rows r+8
#pragma unroll
  for (int t = 0; t < 4; ++t) {
    const int gcol = (cg + t) * 16 + mrow;
    const float bv = bias[gcol];
    float sc = 1.0f, sh = 0.0f;
    if (mode != 0) {
      float s0 = g[gcol] * rsqrtf(v[gcol] + 1e-5f);
      sc = s0;
      sh = be[gcol] - m[gcol] * s0;
    }
#pragma unroll
    for (int r = 0; r < 8; ++r) {
      const int grow = baseRow + rtile * 16 + hi * 8 + r;
      if (grow < N) {
        float tt = acc[t][r] + bv;
        float y;
        if (mode == 0)      y = fmaxf(tt, 0.0f);
        else if (mode == 1) y = fmaxf(tt * sc + sh, 0.0f);
        else                y = fmaxf(tt, 0.0f) * sc + sh;
        out[(size_t)grow * HDIM + gcol] = y;
      }
    }
  }
}

// ---------------------------------------------------------------- pooling
// z[batch[i]][0:128]   += h1[i]; z[batch[i]][128:256] += h2[i]
__global__ __launch_bounds__(256) void pool_add(const float* __restrict__ h1,
                                                const float* __restrict__ h2,
                                                const int* __restrict__ batch,
                                                float* __restrict__ z, int N) {
  int gid = blockIdx.x * 256 + threadIdx.x;
  int i = gid >> 5;
  if (i >= N) return;
  int q = gid & 31;
  int b = batch[i];
  float4 a = ((const float4*)(h1 + (size_t)i * HDIM))[q];
  float4 c = ((const float4*)(h2 + (size_t)i * HDIM))[q];
  float* z1 = z + (size_t)b * 256 + q * 4;
  float* z2 = z1 + 128;
  atomicAdd(z1 + 0, a.x); atomicAdd(z1 + 1, a.y);
  atomicAdd(z1 + 2, a.z); atomicAdd(z1 + 3, a.w);
  atomicAdd(z2 + 0, c.x); atomicAdd(z2 + 1, c.y);
  atomicAdd(z2 + 2, c.z); atomicAdd(z2 + 3, c.w);
}

// ---------------------------------------------------------------- head MLP
// per graph: relu(z[256] @ Wl1[256x64] + bl1) @ Wl2[64x10] + bl2
__global__ __launch_bounds__(64) void head_mlp(const float* __restrict__ z,
                                               const float* __restrict__ Wl1,
                                               const float* __restrict__ bl1,
                                               const float* __restrict__ Wl2,
                                               const float* __restrict__ bl2,
                                               float* __restrict__ out) {
  __shared__ float zs[256];
  __shared__ float hs[64];
  const int gph = blockIdx.x;
  const int t = threadIdx.x;
  for (int i = t; i < 256; i += 64) zs[i] = z[(size_t)gph * 256 + i];
  __syncthreads();
  float acc = bl1[t];
  for (int k = 0; k < 256; ++k) acc = fmaf(zs[k], Wl1[k * 64 + t], acc);
  hs[t] = fmaxf(acc, 0.0f);
  __syncthreads();
  if (t < 10) {
    float o = bl2[t];
    for (int k = 0; k < 64; ++k) o = fmaf(hs[k], Wl2[k * 10 + t], o);
    out[(size_t)gph * 10 + t] = o;
  }
}

// ---------------------------------------------------------------- launch
extern "C" void kernel_launch(void* const* d_in, const int* in_sizes, int n_in,
                              void* d_out, int out_size, void* d_ws, size_t ws_size,
                              hipStream_t stream) {
  const float* x    = (const float*)d_in[0];
  const int*   ei   = (const int*)d_in[1];
  const int*   batch= (const int*)d_in[2];
  const float* W1a  = (const float*)d_in[3];  const float* b1a = (const float*)d_in[4];
  const float* g1   = (const float*)d_in[5];  const float* be1 = (const float*)d_in[6];
  const float* m1   = (const float*)d_in[7];  const float* v1  = (const float*)d_in[8];
  const float* W1b  = (const float*)d_in[9];  const float* b1b = (const float*)d_in[10];
  const float* W2a  = (const float*)d_in[11]; const float* b2a = (const float*)d_in[12];
  const float* g2   = (const float*)d_in[13]; const float* be2 = (const float*)d_in[14];
  const float* m2   = (const float*)d_in[15]; const float* v2  = (const float*)d_in[16];
  const float* W2b  = (const float*)d_in[17]; const float* b2b = (const float*)d_in[18];
  const float* Wl1  = (const float*)d_in[19]; const float* bl1 = (const float*)d_in[20];
  const float* Wl2  = (const float*)d_in[21]; const float* bl2 = (const float*)d_in[22];

  const int N = in_sizes[0] / HDIM;
  const int E = in_sizes[1] / 2;
  const int G = out_size / 10;

  float* bufA = (float*)d_ws;                 // agg buffer; later reused for h2
  float* bufB = bufA + (size_t)N * HDIM;      // MLP intermediate
  float* h1   = bufB + (size_t)N * HDIM;      // persists for pooling
  float* z    = h1 + (size_t)N * HDIM;        // G x 256 pooled
  float* h2   = bufA;

  dim3 blk(256);
  const int n4       = N * 32;
  const int copyGrid = (n4 + 255) / 256;
  const int scatGrid = (E * 32 + 255) / 256;
  const int gemmGrid = (N + 63) / 64;
  const int poolGrid = (N * 32 + 255) / 256;

  // conv1: aggregate then MLP (Linear->BN->ReLU->Linear->ReLU)
  copy_f4<<<copyGrid, blk, 0, stream>>>((const float4*)x, (float4*)bufA, n4);
  scatter_add<<<scatGrid, blk, 0, stream>>>(x, ei, bufA, E);
  gemm128_wmma<<<gemmGrid, blk, 0, stream>>>(bufA, W1a, b1a, g1, be1, m1, v1, bufB, N, 1);
  gemm128_wmma<<<gemmGrid, blk, 0, stream>>>(bufB, W1b, b1b, nullptr, nullptr, nullptr, nullptr, h1, N, 0);

  // conv2: aggregate then MLP (Linear->ReLU->BN->Linear->ReLU)
  copy_f4<<<copyGrid, blk, 0, stream>>>((const float4*)h1, (float4*)bufA, n4);
  scatter_add<<<scatGrid, blk, 0, stream>>>(h1, ei, bufA, E);
  gemm128_wmma<<<gemmGrid, blk, 0, stream>>>(bufA, W2a, b2a, g2, be2, m2, v2, bufB, N, 2);
  gemm128_wmma<<<gemmGrid, blk, 0, stream>>>(bufB, W2b, b2b, nullptr, nullptr, nullptr, nullptr, h2, N, 0);

  // global add pool + head
  zero_f<<<(G * 256 + 255) / 256, blk, 0, stream>>>(z, G * 256);
  pool_add<<<poolGrid, blk, 0, stream>>>(h1, h2, batch, z, N);
  head_mlp<<<G, dim3(64), 0, stream>>>(z, Wl1, bl1, Wl2, bl2, (float*)d_out);
}